// GWConv2d_35244501631226
// MI455X (gfx1250) — compile-verified
//
#include <hip/hip_runtime.h>

typedef __attribute__((ext_vector_type(2))) float v2f;
typedef __attribute__((ext_vector_type(8))) float v8f;

static constexpr int Bn = 8;
static constexpr int HW = 256 * 256;

__device__ __forceinline__ v8f wmma4(v2f a, v2f b, v8f c) {
  // V_WMMA_F32_16X16X4_F32 : D = A(16x4) x B(4x16) + C(16x16), all f32
  return __builtin_amdgcn_wmma_f32_16x16x4_f32(false, a, false, b, (short)0, c,
                                               false, false);
}

__device__ __forceinline__ int br8(int x) {
  return (int)(__brev((unsigned)x) >> 24);
}

// ---------------------------------------------------------------------------
// Weight prep: k_sym = k + conj(flip(k, (p,q))), reordered to [d][o][K=t*64+i]
// ---------------------------------------------------------------------------
__global__ void k_symmetrize(const float* __restrict__ kre,
                             const float* __restrict__ kim,
                             float* __restrict__ wre, float* __restrict__ wim) {
  const int idx = blockIdx.x * 256 + threadIdx.x;
  if (idx >= 4 * 128 * 64 * 9) return;
  const int t = idx % 9;
  const int i = (idx / 9) & 63;
  const int o = (idx / (9 * 64)) % 128;
  const int d = idx / (9 * 64 * 128);
  const int p = t / 3, q = t - p * 3;
  const size_t a = ((((size_t)o * 64 + i) * 3 + p) * 3 + q) * 4 + d;
  const size_t f = ((((size_t)o * 64 + i) * 3 + (2 - p)) * 3 + (2 - q)) * 4 + d;
  const size_t w = ((size_t)(d * 128 + o)) * 576 + t * 64 + i;
  wre[w] = kre[a] + kre[f];
  wim[w] = kim[a] - kim[f];
}

// ---------------------------------------------------------------------------
// Fused 1x1 convs: res -> d_out, h -> hbuf
// ---------------------------------------------------------------------------
__global__ void k_conv1x1_dual(const float* __restrict__ x,
                               const float* __restrict__ cw,
                               const float* __restrict__ cb,
                               const float* __restrict__ ew,
                               const float* __restrict__ eb,
                               float* __restrict__ out,
                               float* __restrict__ hbuf) {
  __shared__ float xs[64 * 65];
  const int bid = blockIdx.x;
  const int tile = bid & 1023;
  const int b = bid >> 10;
  const int pix0 = tile << 6;
  const int tid = threadIdx.x;
  for (int idx = tid; idx < 4096; idx += 256) {
    const int ci = idx >> 6, p = idx & 63;
    xs[ci * 65 + p] = x[((size_t)(b * 64 + ci)) * HW + pix0 + p];
  }
  __syncthreads();
  const int p = tid & 63, ob = (tid >> 6) << 4;
  for (int oo = 0; oo < 16; ++oo) {
    const int o = ob + oo;
    float aR = cb[o], aE = eb[o];
    const float* wr = cw + o * 64;
    const float* we = ew + o * 64;
    for (int i = 0; i < 64; ++i) {
      const float xv = xs[i * 65 + p];
      aR = fmaf(wr[i], xv, aR);
      aE = fmaf(we[i], xv, aE);
    }
    const size_t oi = ((size_t)(b * 64 + o)) * HW + pix0 + p;
    out[oi] = aR;
    hbuf[oi] = aE;
  }
}

// ---------------------------------------------------------------------------
// 256-pt radix-2 DIT FFT along rows, real -> complex
// ---------------------------------------------------------------------------
__global__ void k_fft_row_r2c(const float* __restrict__ h,
                              float2* __restrict__ xrow) {
  __shared__ float sre[256], sim[256];
  const size_t base = (size_t)blockIdx.x * 256;
  const int tid = threadIdx.x;  // 128
  sre[br8(tid)] = h[base + tid];
  sim[br8(tid)] = 0.f;
  sre[br8(tid + 128)] = h[base + tid + 128];
  sim[br8(tid + 128)] = 0.f;
  __syncthreads();
  for (int s = 1; s <= 8; ++s) {
    const int m = 1 << s, hm = m >> 1;
    const int j = tid & (hm - 1);
    const int p1 = ((tid >> (s - 1)) << s) + j;
    const int p2 = p1 + hm;
    const float ang = -6.28318530718f * (float)j / (float)m;
    const float wr = __cosf(ang), wi = __sinf(ang);
    const float xr = sre[p2], xi = sim[p2];
    const float tr = wr * xr - wi * xi, ti = wr * xi + wi * xr;
    const float ur = sre[p1], ui = sim[p1];
    sre[p1] = ur + tr; sim[p1] = ui + ti;
    sre[p2] = ur - tr; sim[p2] = ui - ti;
    __syncthreads();
  }
  xrow[base + tid] = make_float2(sre[tid], sim[tid]);
  xrow[base + tid + 128] = make_float2(sre[tid + 128], sim[tid + 128]);
}

// ---------------------------------------------------------------------------
// Column FFT (16 cols/block) fused with i*f ramps, channel concat, fftshift
// ---------------------------------------------------------------------------
__global__ void k_fft_col_ramp(const float2* __restrict__ xrow,
                               float2* __restrict__ xs) {
  __shared__ float sre[16 * 257], sim[16 * 257];
  const int bid = blockIdx.x;            // b*1024 + c*16 + xt
  const int xt = bid & 15;
  const int c = (bid >> 4) & 63;
  const int b = bid >> 10;
  const int x0 = xt << 4;
  const int tid = threadIdx.x;           // 256
  const size_t ibase = ((size_t)(b * 64 + c)) * HW;
  for (int idx = tid; idx < 4096; idx += 256) {
    const int ky = idx >> 4, xi = idx & 15;
    const float2 v = xrow[ibase + (size_t)ky * 256 + x0 + xi];
    const int kb = br8(ky);
    sre[xi * 257 + kb] = v.x;
    sim[xi * 257 + kb] = v.y;
  }
  __syncthreads();
  const int col = tid & 15, tb = tid >> 4;
  for (int s = 1; s <= 8; ++s) {
    const int m = 1 << s, hm = m >> 1;
    for (int rr = 0; rr < 8; ++rr) {
      const int bf = tb + (rr << 4);
      const int j = bf & (hm - 1);
      const int p1 = col * 257 + ((bf >> (s - 1)) << s) + j;
      const int p2 = p1 + hm;
      const float ang = -6.28318530718f * (float)j / (float)m;
      const float wr = __cosf(ang), wi = __sinf(ang);
      const float xr = sre[p2], xi = sim[p2];
      const float tr = wr * xr - wi * xi, ti = wr * xi + wi * xr;
      const float ur = sre[p1], ui = sim[p1];
      sre[p1] = ur + tr; sim[p1] = ui + ti;
      sre[p2] = ur - tr; sim[p2] = ui - ti;
    }
    __syncthreads();
  }
  for (int idx = tid; idx < 4096; idx += 256) {
    const int ky = idx >> 4, xi = idx & 15;
    const int x = x0 + xi;
    const float Xr = sre[xi * 257 + ky], Xi = sim[xi * 257 + ky];
    const float fh = (float)((ky < 128) ? ky : ky - 256) * (1.0f / 256.0f);
    const float fw = (float)((x < 128) ? x : x - 256) * (1.0f / 256.0f);
    const int ys = (ky + 128) & 255, xsh = (x + 128) & 255;
    const size_t sp = (size_t)ys * 256 + xsh;
    xs[((size_t)(b * 128 + c)) * HW + sp] = make_float2(-fh * Xi, fh * Xr);
    xs[((size_t)(b * 128 + 64 + c)) * HW + sp] = make_float2(-fw * Xi, fw * Xr);
  }
}

// ---------------------------------------------------------------------------
// Grouped complex 3x3 conv layer via f32 WMMA implicit GEMM.
// K = tap*64 + in_ch (tap-major -> every K=4 step stays inside one tap).
// Karatsuba complex: m1=kr*xr, m2=ki*xi, m3=(kr+ki)(xr+xi);
//                    re = m1-m2, im = m3-m1-m2  => 3 WMMAs per K-step.
// ---------------------------------------------------------------------------
__global__ void k_spectral_conv(const float2* __restrict__ src,
                                float2* __restrict__ dst,
                                const float* __restrict__ wre,
                                const float* __restrict__ wim, int d) {
  __shared__ float sre[3 * 34 * 68];  // [row][col][ch], ch padded 64->68
  __shared__ float sim[3 * 34 * 68];
  const int bid = blockIdx.x;           // b*4096 + g*2048 + y*8 + xt
  const int xt = bid & 7;
  const int y = (bid >> 3) & 255;
  const int g = (bid >> 11) & 1;
  const int b = bid >> 12;
  const int x0 = xt << 5;
  const int tid = threadIdx.x;          // 256 = 8 waves

  for (int idx = tid; idx < 3 * 34 * 64; idx += 256) {
    const int cc = idx % 34;
    const int tmp = idx / 34;
    const int r = tmp % 3;
    const int i = tmp / 3;
    const int yy = y - 1 + r;
    const int xx = x0 - 1 + cc;
    float2 v = make_float2(0.f, 0.f);
    if (yy >= 0 && yy < 256 && xx >= 0 && xx < 256)
      v = src[((size_t)(b * 128 + g * 64 + i)) * HW + (size_t)yy * 256 + xx];
    const int si = (r * 34 + cc) * 68 + i;
    sre[si] = v.x;
    sim[si] = v.y;
  }
  __syncthreads();

  const int wave = tid >> 5, lane = tid & 31;
  const int half = lane >> 4, lp = lane & 15;
  const int m0 = (wave & 3) << 4;       // out-channel tile within group
  const int n0 = (wave >> 2) << 4;      // pixel tile within 32-wide strip
  // per-lane base pointers; K-loop then uses pure immediate-style offsets
  const int wbase = (d * 128 + g * 64 + m0 + lp) * 576 + 2 * half;
  const float* __restrict__ wpr = wre + wbase;
  const float* __restrict__ wpi = wim + wbase;

  v8f accM1 = {};  // sum kr*xr
  v8f accM2 = {};  // sum ki*xi
  v8f accM3 = {};  // sum (kr+ki)*(xr+xi)
  for (int t = 0; t < 9; ++t) {
    const int p = t / 3, q = t - p * 3;
    const int crow = (p * 34 + n0 + lp + q) * 68 + 2 * half;
    const int kb = t * 64;
#pragma unroll 8
    for (int i0 = 0; i0 < 64; i0 += 4) {
      v2f akr; akr.x = wpr[kb + i0]; akr.y = wpr[kb + i0 + 1];
      v2f aki; aki.x = wpi[kb + i0]; aki.y = wpi[kb + i0 + 1];
      const int si = crow + i0;
      v2f pr; pr.x = sre[si]; pr.y = sre[si + 1];
      v2f pi; pi.x = sim[si]; pi.y = sim[si + 1];
      const v2f aks = akr + aki;   // VALU, co-executes with matrix pipe
      const v2f ps = pr + pi;
      accM1 = wmma4(akr, pr, accM1);
      accM2 = wmma4(aki, pi, accM2);
      accM3 = wmma4(aks, ps, accM3);
    }
  }

  const int xx = x0 + n0 + lp;
  for (int r = 0; r < 8; ++r) {
    const int o = g * 64 + m0 + r + (half << 3);
    const float m1 = accM1[r], m2 = accM2[r], m3 = accM3[r];
    dst[((size_t)(b * 128 + o)) * HW + (size_t)y * 256 + xx] =
        make_float2(m1 - m2, m3 - m1 - m2);
  }
}

// ---------------------------------------------------------------------------
// Inverse column FFT fused with ifftshift (shift by 128 == its own inverse)
// ---------------------------------------------------------------------------
__global__ void k_ifft_col_unshift(const float2* __restrict__ xs,
                                   float2* __restrict__ y1) {
  __shared__ float sre[16 * 257], sim[16 * 257];
  const int bid = blockIdx.x;            // b*2048 + c*16 + xt
  const int xt = bid & 15;
  const int c = (bid >> 4) & 127;
  const int b = bid >> 11;
  const int x0 = xt << 4;
  const int tid = threadIdx.x;
  const size_t ibase = ((size_t)(b * 128 + c)) * HW;
  for (int idx = tid; idx < 4096; idx += 256) {
    const int ky = idx >> 4, xi = idx & 15;
    const int srow = (ky + 128) & 255;
    const int scol = (x0 + xi + 128) & 255;
    const float2 v = xs[ibase + (size_t)srow * 256 + scol];
    const int kb = br8(ky);
    sre[xi * 257 + kb] = v.x;
    sim[xi * 257 + kb] = v.y;
  }
  __syncthreads();
  const int col = tid & 15, tb = tid >> 4;
  for (int s = 1; s <= 8; ++s) {
    const int m = 1 << s, hm = m >> 1;
    for (int rr = 0; rr < 8; ++rr) {
      const int bf = tb + (rr << 4);
      const int j = bf & (hm - 1);
      const int p1 = col * 257 + ((bf >> (s - 1)) << s) + j;
      const int p2 = p1 + hm;
      const float ang = 6.28318530718f * (float)j / (float)m;
      const float wr = __cosf(ang), wi = __sinf(ang);
      const float xr = sre[p2], xi = sim[p2];
      const float tr = wr * xr - wi * xi, ti = wr * xi + wi * xr;
      const float ur = sre[p1], ui = sim[p1];
      sre[p1] = ur + tr; sim[p1] = ui + ti;
      sre[p2] = ur - tr; sim[p2] = ui - ti;
    }
    __syncthreads();
  }
  const float sc = 1.0f / 256.0f;
  for (int idx = tid; idx < 4096; idx += 256) {
    const int yy = idx >> 4, xi = idx & 15;
    y1[ibase + (size_t)yy * 256 + x0 + xi] =
        make_float2(sre[xi * 257 + yy] * sc, sim[xi * 257 + yy] * sc);
  }
}

// ---------------------------------------------------------------------------
// Inverse row FFT, keep real part (with 1/256 scale)
// ---------------------------------------------------------------------------
__global__ void k_ifft_row_real(const float2* __restrict__ y1,
                                float* __restrict__ yreal) {
  __shared__ float sre[256], sim[256];
  const size_t base = (size_t)blockIdx.x * 256;
  const int tid = threadIdx.x;  // 128
  {
    const float2 a = y1[base + tid];
    const float2 b = y1[base + tid + 128];
    sre[br8(tid)] = a.x; sim[br8(tid)] = a.y;
    sre[br8(tid + 128)] = b.x; sim[br8(tid + 128)] = b.y;
  }
  __syncthreads();
  for (int s = 1; s <= 8; ++s) {
    const int m = 1 << s, hm = m >> 1;
    const int j = tid & (hm - 1);
    const int p1 = ((tid >> (s - 1)) << s) + j;
    const int p2 = p1 + hm;
    const float ang = 6.28318530718f * (float)j / (float)m;
    const float wr = __cosf(ang), wi = __sinf(ang);
    const float xr = sre[p2], xi = sim[p2];
    const float tr = wr * xr - wi * xi, ti = wr * xi + wi * xr;
    const float ur = sre[p1], ui = sim[p1];
    sre[p1] = ur + tr; sim[p1] = ui + ti;
    sre[p2] = ur - tr; sim[p2] = ui - ti;
    __syncthreads();
  }
  const float sc = 1.0f / 256.0f;
  yreal[base + tid] = sre[tid] * sc;
  yreal[base + tid + 128] = sre[tid + 128] * sc;
}

// ---------------------------------------------------------------------------
// proj 1x1 (128->64) and out += 0.001 * y
// ---------------------------------------------------------------------------
__global__ void k_proj_axpy(const float* __restrict__ yreal,
                            const float* __restrict__ pw,
                            const float* __restrict__ pb,
                            float* __restrict__ out) {
  __shared__ float ys[128 * 65];
  const int bid = blockIdx.x;
  const int tile = bid & 1023;
  const int b = bid >> 10;
  const int pix0 = tile << 6;
  const int tid = threadIdx.x;
  for (int idx = tid; idx < 8192; idx += 256) {
    const int ci = idx >> 6, p = idx & 63;
    ys[ci * 65 + p] = yreal[((size_t)(b * 128 + ci)) * HW + pix0 + p];
  }
  __syncthreads();
  const int p = tid & 63, ob = (tid >> 6) << 4;
  for (int oo = 0; oo < 16; ++oo) {
    const int o = ob + oo;
    float acc = pb[o];
    const float* w = pw + o * 128;
    for (int i = 0; i < 128; ++i) acc = fmaf(w[i], ys[i * 65 + p], acc);
    const size_t oi = ((size_t)(b * 64 + o)) * HW + pix0 + p;
    out[oi] += 0.001f * acc;
  }
}

// ---------------------------------------------------------------------------
extern "C" void kernel_launch(void* const* d_in, const int* in_sizes, int n_in,
                              void* d_out, int out_size, void* d_ws,
                              size_t ws_size, hipStream_t stream) {
  (void)in_sizes; (void)n_in; (void)out_size; (void)ws_size;
  const float* x      = (const float*)d_in[0];
  const float* conv_w = (const float*)d_in[1];
  const float* conv_b = (const float*)d_in[2];
  const float* emb_w  = (const float*)d_in[3];
  const float* emb_b  = (const float*)d_in[4];
  const float* proj_w = (const float*)d_in[5];
  const float* proj_b = (const float*)d_in[6];
  const float* kre    = (const float*)d_in[7];
  const float* kim    = (const float*)d_in[8];
  float* out = (float*)d_out;
  char* ws = (char*)d_ws;

  const size_t SZ_W   = (size_t)4 * 128 * 576 * 4;       // 1.18 MB
  const size_t SZ_H   = (size_t)Bn * 64 * HW * 4;        // 134 MB
  const size_t SZ_XR  = (size_t)Bn * 64 * HW * 8;        // 268 MB
  const size_t SZ_BIG = (size_t)Bn * 128 * HW * 8;       // 536 MB
  float*  wre  = (float*)(ws);
  float*  wim  = (float*)(ws + SZ_W);
  float*  hbuf = (float*)(ws + 2 * SZ_W);
  float2* xrow = (float2*)(ws + 2 * SZ_W + SZ_H);
  float2* XA   = (float2*)(ws + 2 * SZ_W + SZ_H + SZ_XR);
  float2* XB   = (float2*)(ws + 2 * SZ_W + SZ_H + SZ_XR + SZ_BIG);
  float*  yreal = (float*)xrow;  // xrow dead after k_fft_col_ramp

  k_symmetrize<<<1152, 256, 0, stream>>>(kre, kim, wre, wim);
  k_conv1x1_dual<<<Bn * 1024, 256, 0, stream>>>(x, conv_w, conv_b, emb_w,
                                                emb_b, out, hbuf);
  k_fft_row_r2c<<<Bn * 64 * 256, 128, 0, stream>>>(hbuf, xrow);
  k_fft_col_ramp<<<Bn * 64 * 16, 256, 0, stream>>>(xrow, XA);

  const float2* cin = XA;
  float2* cout = XB;
  for (int d = 0; d < 4; ++d) {
    k_spectral_conv<<<Bn * 2 * 256 * 8, 256, 0, stream>>>(cin, cout, wre, wim,
                                                          d);
    float2* t = (float2*)cin; cin = cout; cout = t;
  }
  // after 4 layers result is back in XA (cin == XA)
  k_ifft_col_unshift<<<Bn * 128 * 16, 256, 0, stream>>>(cin, XB);
  k_ifft_row_real<<<Bn * 128 * 256, 128, 0, stream>>>(XB, yreal);
  k_proj_axpy<<<Bn * 1024, 256, 0, stream>>>(yreal, proj_w, proj_b, out);
}